// FittingModel_76227079570086
// MI455X (gfx1250) — compile-verified
//
#include <hip/hip_runtime.h>
#include <stdint.h>

#define EPSV 0.0001f

typedef __attribute__((ext_vector_type(2))) float v2f;
typedef __attribute__((ext_vector_type(8))) float v8f;

// ---------------------------------------------------------------------------
// Elementwise helpers
// ---------------------------------------------------------------------------

__global__ void k_bnrelu(const float* __restrict__ x, float* __restrict__ y,
                         const float* __restrict__ g, const float* __restrict__ b,
                         const float* __restrict__ m, const float* __restrict__ v,
                         int N, int C) {
  int t = blockIdx.x * blockDim.x + threadIdx.x;
  if (t >= N * C) return;
  int j = t % C;
  float s   = g[j] * rsqrtf(v[j] + EPSV);
  float val = (x[t] - m[j]) * s + b[j];
  y[t] = val > 0.f ? val : 0.f;
}

__global__ void k_zero(float* __restrict__ p, int n) {
  int t = blockIdx.x * blockDim.x + threadIdx.x;
  if (t < n) p[t] = 0.f;
}

// Level-0 conv: Cin=3 -> Cout=32 (tiny fraction of total FLOPs, scalar is fine)
__global__ void k_gconv3(const float* __restrict__ colors, const int* __restrict__ rb,
                         const float* __restrict__ W0, float* __restrict__ out, int N) {
  int t = blockIdx.x * blockDim.x + threadIdx.x;
  if (t >= N * 32) return;
  int i = t >> 5, co = t & 31;
  float acc = 0.f;
  for (int k = 0; k < 27; ++k) {
    int idx = rb[(size_t)i * 27 + k];
    if (idx < N) {
      float c0 = colors[(size_t)idx * 3 + 0];
      float c1 = colors[(size_t)idx * 3 + 1];
      float c2 = colors[(size_t)idx * 3 + 2];
      acc += c0 * W0[(k * 3 + 0) * 32 + co];
      acc += c1 * W0[(k * 3 + 1) * 32 + co];
      acc += c2 * W0[(k * 3 + 2) * 32 + co];
    }
  }
  out[t] = acc;
}

__global__ void k_maxpool(const float* __restrict__ x0, const int* __restrict__ rb,
                          float* __restrict__ out, int Nout, int Nin) {
  int t = blockIdx.x * blockDim.x + threadIdx.x;
  if (t >= Nout * 32) return;
  int i = t >> 5, c = t & 31;
  float acc = -__builtin_inff();
  for (int k = 0; k < 27; ++k) {
    int idx = rb[(size_t)i * 27 + k];
    if (idx < Nin) acc = fmaxf(acc, x0[(size_t)idx * 32 + c]);
  }
  out[t] = acc;
}

// Final head: (1 x 192) @ (192 x 6) + b
__global__ void k_head(const float* __restrict__ t, const float* __restrict__ Wp,
                       const float* __restrict__ bp, float* __restrict__ out) {
  int j = threadIdx.x;
  if (j < 6) {
    float acc = bp[j];
    for (int c = 0; c < 192; ++c) acc += t[c] * Wp[c * 6 + j];
    out[j] = acc;
  }
}

// ---------------------------------------------------------------------------
// WMMA gather-GEMM:  out[i,:] (+)= sum_k  xp[rb[i,k], :] @ W[k,:,:]
//   xp : (Npad+1) x Cin  f32, row Npad is all-zero (padding target)
//   rb : Nout x 27 int32, value Npad means "no neighbor"
//   W  : 27 x Cin x Cout f32
//
// 256 threads = 8 waves; each wave owns a 16-row M-tile and all NT Cout-tiles.
// Accumulators live in VGPRs across the whole 27*Cin reduction (one store pass).
// Weight slabs (16 x Cout of W[k]) are DOUBLE-BUFFERED in LDS and filled with
// gfx1250 async loads (global_load_async_to_lds_b128, ASYNCcnt): while the
// waves run WMMA on slab s, the async engine fills slab s+1. Per-wave
// s_wait_asynccnt retires the oldest slab in-order; a block barrier then makes
// all waves' portions visible before the ds reads.
// A fragment (f32 16x4): lanes 0-15 hold K=0,1 of row M=lane; lanes 16-31 K=2,3.
// B fragment mirrors A with N across lanes.
// ---------------------------------------------------------------------------
template <int NT, bool ACC>
__launch_bounds__(256, 1)
__global__ void k_gconv_wmma(const float* __restrict__ xp, const int* __restrict__ rb,
                             const float* __restrict__ W, float* __restrict__ out,
                             int Nout, int Npad, int Cin) {
  constexpr int Cout   = NT * 16;
  constexpr int NITEMS = 4 * Cout;                 // float4 items per 16 x Cout slab
  constexpr int NITER  = (NITEMS + 255) / 256;     // async issues per thread per slab
  __shared__ __align__(16) float Bs[2][16 * Cout];

  const int tid  = threadIdx.x;
  const int wave = tid >> 5;
  const int lane = tid & 31;
  const int nl   = lane & 15;   // M for A, N for B/C/D
  const int kh   = lane >> 4;   // K-half select
  const int rowbase = (blockIdx.x * 8 + wave) * 16;
  const int myrow   = rowbase + nl;

  v8f vzero = {};
  v8f acc[NT];
#pragma unroll
  for (int nt = 0; nt < NT; ++nt) {
    if (ACC) {
#pragma unroll
      for (int r = 0; r < 8; ++r) {
        int row = rowbase + r + 8 * kh;
        int rc  = row < Nout ? row : 0;
        float vld = out[(size_t)rc * Cout + nt * 16 + nl];
        acc[nt][r] = (row < Nout) ? vld : 0.f;
      }
    } else {
      acc[nt] = vzero;
    }
  }

  // Async fill of one 16 x Cout weight slab into LDS buffer `buf`.
  auto issue_slab = [&](int k, int cg, int buf) {
    const float4* __restrict__ Wk = (const float4*)(W + ((size_t)k * Cin + cg) * Cout);
    unsigned lds0 = (unsigned)(uintptr_t)(&Bs[buf][0]);
#pragma unroll
    for (int it = 0; it < NITER; ++it) {
      int idx = tid + it * 256;
      idx = idx < NITEMS ? idx : NITEMS - 1;       // clamp: duplicate writes are benign
      uint64_t ga = (uint64_t)(uintptr_t)(Wk + idx);
      unsigned  la = lds0 + (unsigned)idx * 16u;
      asm volatile("global_load_async_to_lds_b128 %0, %1, off"
                   :: "v"(la), "v"(ga) : "memory");
    }
  };

  const int GPK = Cin >> 4;                        // 16-row groups per kernel offset
  const int S   = 27 * GPK;                        // total stages
  int kI = 0, cgI = 0;                             // next stage to issue
  int kC = 0, cgC = 0;                             // stage to consume

  issue_slab(kI, cgI, 0);
  cgI += 16; if (cgI == Cin) { cgI = 0; ++kI; }

  const float* __restrict__ arow = xp;             // set when cgC == 0
  for (int s = 0; s < S; ++s) {
    const int cur = s & 1;
    if (s + 1 < S) {
      issue_slab(kI, cgI, cur ^ 1);
      cgI += 16; if (cgI == Cin) { cgI = 0; ++kI; }
      asm volatile("s_wait_asynccnt %0" :: "i"(NITER) : "memory");
    } else {
      asm volatile("s_wait_asynccnt 0" ::: "memory");
    }
    __syncthreads();                               // slab `cur` visible to all waves

    if (cgC == 0) {
      // New kernel offset: per-row gather index (selects keep EXEC all-ones)
      int rrow = myrow < Nout ? myrow : Nout - 1;
      int ri   = rb[(size_t)rrow * 27 + kC];
      ri = (myrow < Nout) ? ri : Npad;             // pad row (zeros) for tail rows
      arow = xp + (size_t)ri * Cin;
      __builtin_prefetch(arow, 0, 1);              // global_prefetch_b8
    }

    const float* __restrict__ Bc = &Bs[cur][0];
#pragma unroll
    for (int kc = 0; kc < 4; ++kc) {
      const int cb = cgC + 4 * kc;
      v2f a = *(const v2f*)(arow + cb + 2 * kh);   // 8B gathered A fragment
#pragma unroll
      for (int nt = 0; nt < NT; ++nt) {
        v2f b;
        b.x = Bc[(4 * kc + 2 * kh + 0) * Cout + nt * 16 + nl];
        b.y = Bc[(4 * kc + 2 * kh + 1) * Cout + nt * 16 + nl];
        acc[nt] = __builtin_amdgcn_wmma_f32_16x16x4_f32(
            false, a, false, b, (short)0, acc[nt], false, false);
      }
    }
    cgC += 16; if (cgC == Cin) { cgC = 0; ++kC; }
    __syncthreads();                               // all reads done before slab reuse
  }

  // Store (divergence only here, after all WMMAs)
#pragma unroll
  for (int nt = 0; nt < NT; ++nt) {
#pragma unroll
    for (int r = 0; r < 8; ++r) {
      int row = rowbase + r + 8 * kh;
      if (row < Nout) out[(size_t)row * Cout + nt * 16 + nl] = acc[nt][r];
    }
  }
}

// ---------------------------------------------------------------------------
// Host orchestration
// ---------------------------------------------------------------------------

static inline unsigned cdiv(size_t a, size_t b) { return (unsigned)((a + b - 1) / b); }

static void launch_gconv(int Cout, bool accum, const float* xp, const int* rb,
                         const float* W, float* out, int Nout, int Npad, int Cin,
                         hipStream_t s) {
  dim3 grid(cdiv((size_t)Nout, 128)), block(256);
#define GC_CASE(NT)                                                                     \
  if (accum) k_gconv_wmma<NT, true><<<grid, block, 0, s>>>(xp, rb, W, out, Nout, Npad, Cin); \
  else       k_gconv_wmma<NT, false><<<grid, block, 0, s>>>(xp, rb, W, out, Nout, Npad, Cin);
  switch (Cout / 16) {
    case 2:  { GC_CASE(2)  } break;
    case 4:  { GC_CASE(4)  } break;
    case 6:  { GC_CASE(6)  } break;
    case 8:  { GC_CASE(8)  } break;
    case 10: { GC_CASE(10) } break;
    case 12: { GC_CASE(12) } break;
    default: break;
  }
#undef GC_CASE
}

static void launch_bnrelu(const float* x, float* y, const float* g, const float* b,
                          const float* m, const float* v, int N, int C, hipStream_t s) {
  size_t nel = (size_t)N * C;
  k_bnrelu<<<cdiv(nel, 256), 256, 0, s>>>(x, y, g, b, m, v, N, C);
}

extern "C" void kernel_launch(void* const* d_in, const int* in_sizes, int n_in,
                              void* d_out, int out_size, void* d_ws, size_t ws_size,
                              hipStream_t stream) {
  // Input flattening (setup_inputs insertion order):
  //  0 coords | 1 colors | 2 W0
  //  3 + 21*b : block b = bn_a(4), W_a, bn_b(4), W_b, W_res, bn_c(4), W_c, bn_d(4), W_d
  //  129..132 bn_final | 133 W_pred | 134 b_pred
  //  135 sub0 | 136 pool | 137+2b down_b | 138+2b sub_b
  (void)n_in; (void)ws_size; (void)out_size;
  const int CH[6] = {32, 64, 96, 128, 160, 192};
  auto F = [&](int i) { return (const float*)d_in[i]; };
  auto I = [&](int i) { return (const int*)d_in[i]; };

  const float* colors = F(1);
  const int    N0     = in_sizes[1] / 3;
  const float* W0     = F(2);
  const float* Wpred  = F(133);
  const float* bpred  = F(134);
  const int*   rb0    = I(135);
  const int*   rbpool = I(136);

  int Nlev[8];
  Nlev[0] = N0;
  Nlev[1] = in_sizes[136] / 27;
  for (int b = 0; b < 6; ++b) Nlev[b + 2] = in_sizes[137 + 2 * b] / 27;
  int clev[8];
  clev[0] = 32; clev[1] = 32;
  for (int b = 0; b < 6; ++b) clev[b + 2] = CH[b];

  // Workspace arenas (floats): x0 plus 3 ping-pong buffers sized for the
  // largest (N+1) x C level. All levels fit comfortably in the 192MB L2.
  float* ws  = (float*)d_ws;
  size_t off = ((size_t)N0 * 32 + 3) & ~(size_t)3;
  float* x0  = ws;
  size_t maxEl = 0;
  for (int L = 1; L <= 7; ++L) {
    size_t e = (size_t)(Nlev[L] + 1) * clev[L];
    if (e > maxEl) maxEl = e;
  }
  maxEl = (maxEl + 3) & ~(size_t)3;
  float* buf0 = ws + off;
  float* buf1 = ws + off + maxEl;
  float* buf2 = ws + off + 2 * maxEl;

  // Level 0: conv(3->32) then maxpool to level 1
  k_gconv3<<<cdiv((size_t)N0 * 32, 256), 256, 0, stream>>>(colors, rb0, W0, x0, N0);
  float* X = buf0;  float* S = buf1;  float* T = buf2;
  k_maxpool<<<cdiv((size_t)Nlev[1] * 32, 256), 256, 0, stream>>>(x0, rbpool, X, Nlev[1], N0);
  k_zero<<<1, 256, 0, stream>>>(X + (size_t)Nlev[1] * 32, 32);  // pad row

  int Nin = Nlev[1], cin = 32;
  for (int b = 0; b < 6; ++b) {
    const int base = 3 + b * 21;
    const float *ga = F(base + 0),  *ba = F(base + 1),  *ma = F(base + 2),  *va = F(base + 3);
    const float* Wa   = F(base + 4);
    const float *gb = F(base + 5),  *bb = F(base + 6),  *mb = F(base + 7),  *vb = F(base + 8);
    const float* Wb   = F(base + 9);
    const float* Wres = F(base + 10);
    const float *gc = F(base + 11), *bc = F(base + 12), *mc = F(base + 13), *vc = F(base + 14);
    const float* Wc   = F(base + 15);
    const float *gd = F(base + 16), *bd = F(base + 17), *md = F(base + 18), *vd = F(base + 19);
    const float* Wd   = F(base + 20);
    const int* rbdown = I(137 + 2 * b);
    const int* rbsub  = I(138 + 2 * b);
    const int Nout = Nlev[b + 2], cout = CH[b];

    // a = gconv(bnrelu(x), down, W_a)
    launch_bnrelu(X, T, ga, ba, ma, va, Nin, cin, stream);
    k_zero<<<1, 256, 0, stream>>>(T + (size_t)Nin * cin, cin);
    launch_gconv(cout, false, T, rbdown, Wa, S, Nout, Nin, cin, stream);

    // a = gconv(bnrelu(a), sub, W_b)
    launch_bnrelu(S, T, gb, bb, mb, vb, Nout, cout, stream);
    k_zero<<<1, 256, 0, stream>>>(T + (size_t)Nout * cout, cout);
    launch_gconv(cout, false, T, rbsub, Wb, S, Nout, Nout, cout, stream);

    // x = a + gconv(x, down, W_res)   (fused: accumulate into S)
    launch_gconv(cout, true, X, rbdown, Wres, S, Nout, Nin, cin, stream);
    k_zero<<<1, 256, 0, stream>>>(S + (size_t)Nout * cout, cout);  // pad row of new x
    { float* t = X; X = S; S = t; }  // new x lives in X

    // a = gconv(bnrelu(x), sub, W_c)
    launch_bnrelu(X, T, gc, bc, mc, vc, Nout, cout, stream);
    k_zero<<<1, 256, 0, stream>>>(T + (size_t)Nout * cout, cout);
    launch_gconv(cout, false, T, rbsub, Wc, S, Nout, Nout, cout, stream);

    // x = x + gconv(bnrelu(a), sub, W_d)   (fused: accumulate into X)
    launch_bnrelu(S, T, gd, bd, md, vd, Nout, cout, stream);
    k_zero<<<1, 256, 0, stream>>>(T + (size_t)Nout * cout, cout);
    launch_gconv(cout, true, T, rbsub, Wd, X, Nout, Nout, cout, stream);

    Nin = Nout; cin = cout;
  }

  // Head: bnrelu(x, bn_final) @ W_pred + b_pred  (Nin == 1 at this point)
  launch_bnrelu(X, T, F(129), F(130), F(131), F(132), Nin, 192, stream);
  k_head<<<1, 32, 0, stream>>>(T, Wpred, bpred, (float*)d_out);
}